// QConv2d_37709812859575
// MI455X (gfx1250) — compile-verified
//
#include <hip/hip_runtime.h>
#include <math.h>

// CDNA5 / gfx1250, wave32. f32 WMMA: D(16x16) = A(16x4) * B(4x16) + C.
// Operand lane maps (ISA 7.12.2):
//   A: row = lane&15, K = 4*kb + 2*(lane>>4) (+1 in second VGPR)
//   B: N   = lane&15, K = 4*kb + 2*(lane>>4) (+1 in second VGPR)
//   C/D: row = r + 8*(lane>>4), N = lane&15
typedef float v2f __attribute__((ext_vector_type(2)));
typedef float v8f __attribute__((ext_vector_type(8)));

__device__ __forceinline__ v8f wmma_f32_16x16x4(v2f a, v2f b, v8f c) {
  return __builtin_amdgcn_wmma_f32_16x16x4_f32(false, a, false, b,
                                               (short)0, c, false, false);
}

// ---------------------------------------------------------------------------
// Build the two 32x32 block-diagonal register unitaries from the 28 angles.
// ws layout: [0..1024) = Ux (row-major 32x32), [1024..2048) = Uy.
// Note: the top-left 16x16 tile of each equals diag(Uf, Uf) — that is the
// only A/B operand tile the transform kernels ever need.
// ---------------------------------------------------------------------------
__global__ void build_unitaries(const float* __restrict__ phi_x,
                                const float* __restrict__ phi_y,
                                float* __restrict__ Ubuf) {
  __shared__ float Uf[2][8][8];
  const int t = threadIdx.x;
  if (t < 2) {
    const float* phi = (t == 0) ? phi_x : phi_y;
    float U[8][8];
    for (int i = 0; i < 8; ++i)
      for (int j = 0; j < 8; ++j) U[i][j] = (i == j) ? 1.0f : 0.0f;
    int idx = 0;
    for (int i = 1; i < 8; ++i) {
      for (int j = i; j > 0; --j) {
        const float c = cosf(phi[idx]);
        const float s = sinf(phi[idx]);
        for (int q = 0; q < 8; ++q) {
          const float r0 = c * U[j - 1][q] + s * U[j][q];
          const float r1 = -s * U[j - 1][q] + c * U[j][q];
          U[j - 1][q] = r0;
          U[j][q] = r1;
        }
        ++idx;
      }
    }
    for (int i = 0; i < 8; ++i)
      for (int j = 0; j < 8; ++j) Uf[t][i][j] = U[i][j];
  }
  __syncthreads();
  for (int e = t; e < 2048; e += blockDim.x) {
    const int which = e >> 10;
    const int rc = e & 1023;
    const int r = rc >> 5, cc = rc & 31;
    float v = 0.0f;
    if ((r >> 3) == (cc >> 3)) v = Uf[which][r & 7][cc & 7];
    Ubuf[e] = v;
  }
}

// ---------------------------------------------------------------------------
// Fused left conjugation: Y[b, ix, iy, :] = sum_{jx,jy} Ux[ix,jx] Uy[iy,jy]
// X[b, jx, jy, :].  Mixing happens only inside (8-block jx) x (8-block jy)
// = 64-row groups.  One workgroup: (batch, jx-block, jy-block, 64-col chunk),
// 64x64 f32 tile staged in LDS (row stride 68: 16B-aligned, bank-rotated).
// Stage A contracts jy (consecutive-row tiles); stage B contracts jx using
// stride-8 LDS row addressing.  A-operands = diag(Uf,Uf) = tile0 of Ux/Uy.
// Every 16x16 patch is read and written by exactly one wave -> LDS in-place.
// ---------------------------------------------------------------------------
__global__ void __launch_bounds__(256)
fused_left(const float* __restrict__ X, float* __restrict__ Y,
           const float* __restrict__ Uxm, const float* __restrict__ Uym) {
  __shared__ float T[64 * 68];
  const int t = threadIdx.x;
  const int lane = t & 31;
  const int wave = t >> 5;
  const int row16 = lane & 15;
  const int hi = lane >> 4;

  const int cc = blockIdx.x & 15;         // 64-wide column chunk
  const int b2 = (blockIdx.x >> 4) & 3;   // jy block
  const int a = (blockIdx.x >> 6) & 3;    // jx block
  const int b = blockIdx.x >> 8;          // batch

  const long long base = (long long)b * 1048576 + (long long)cc * 64;

  // Load 64 rows x 64 cols (float4, coalesced).
  for (int e4 = t; e4 < 1024; e4 += 256) {
    const int lr = e4 >> 4;
    const int c4 = e4 & 15;
    const int grow = (8 * a + (lr >> 3)) * 32 + 8 * b2 + (lr & 7);
    const float4 v =
        *(const float4*)(X + base + (long long)grow * 1024 + c4 * 4);
    *(float4*)(&T[lr * 68 + c4 * 4]) = v;
  }
  __syncthreads();

  v2f aY[4], aX[4];
#pragma unroll
  for (int kb = 0; kb < 4; ++kb) {
    const int k = 4 * kb + 2 * hi;
    aY[kb].x = Uym[row16 * 32 + k];
    aY[kb].y = Uym[row16 * 32 + k + 1];
    aX[kb].x = Uxm[row16 * 32 + k];
    aX[kb].y = Uxm[row16 * 32 + k + 1];
  }

  // Stage A: contract jy.  16 tasks = 4 row-tiles x 4 col-tiles, disjoint.
#pragma unroll
  for (int it = 0; it < 2; ++it) {
    const int task = wave + 8 * it;
    const int t4 = task >> 2;
    const int n0 = (task & 3) * 16;
    v8f c = {};
#pragma unroll
    for (int kb = 0; kb < 4; ++kb) {
      const int k = 4 * kb + 2 * hi;
      v2f bv;
      bv.x = T[(16 * t4 + k) * 68 + n0 + row16];
      bv.y = T[(16 * t4 + k + 1) * 68 + n0 + row16];
      c = wmma_f32_16x16x4(aY[kb], bv, c);
    }
#pragma unroll
    for (int r = 0; r < 8; ++r)
      T[(16 * t4 + r + 8 * hi) * 68 + n0 + row16] = c[r];
  }
  __syncthreads();

  // Stage B: contract jx.  Tile row i <-> LDS row (i&7)*8 + 2*t4 + (i>>3).
#pragma unroll
  for (int it = 0; it < 2; ++it) {
    const int task = wave + 8 * it;
    const int t4 = task >> 2;
    const int n0 = (task & 3) * 16;
    v8f c = {};
#pragma unroll
    for (int kb = 0; kb < 4; ++kb) {
      const int k0 = 4 * kb + 2 * hi;
      const int k1 = k0 + 1;
      const int lr0 = (k0 & 7) * 8 + 2 * t4 + (k0 >> 3);
      const int lr1 = (k1 & 7) * 8 + 2 * t4 + (k1 >> 3);
      v2f bv;
      bv.x = T[lr0 * 68 + n0 + row16];
      bv.y = T[lr1 * 68 + n0 + row16];
      c = wmma_f32_16x16x4(aX[kb], bv, c);
    }
#pragma unroll
    for (int r = 0; r < 8; ++r)
      T[(r * 8 + 2 * t4 + hi) * 68 + n0 + row16] = c[r];
  }
  __syncthreads();

  // Store back (float4, coalesced).
  for (int e4 = t; e4 < 1024; e4 += 256) {
    const int lr = e4 >> 4;
    const int c4 = e4 & 15;
    const int grow = (8 * a + (lr >> 3)) * 32 + 8 * b2 + (lr & 7);
    *(float4*)(Y + base + (long long)grow * 1024 + c4 * 4) =
        *(const float4*)(&T[lr * 68 + c4 * 4]);
  }
}

// ---------------------------------------------------------------------------
// Fused right conjugation (in place): for each of the 16384 rows, its 32x32
// (kx,ky) matrix W (1024 contiguous floats) becomes Ux-apply(kx) and
// Uy-apply(ky) of W.  Two rows per workgroup; LDS row stride 33 (kills the
// 8-way bank conflict of the kx-strided reads).  8 waves = 8 disjoint
// 16x16 tile tasks per stage.
// ---------------------------------------------------------------------------
__global__ void __launch_bounds__(256)
fused_right(float* __restrict__ buf, const float* __restrict__ Uxm,
            const float* __restrict__ Uym) {
  __shared__ float T[2][32 * 33];
  const int t = threadIdx.x;
  const int lane = t & 31;
  const int wave = t >> 5;
  const int row16 = lane & 15;
  const int hi = lane >> 4;
  const long long m0 = (long long)blockIdx.x * 2;

  for (int e = t; e < 2048; e += 256) {
    const int i = e >> 10;
    const int e1 = e & 1023;
    T[i][(e1 >> 5) * 33 + (e1 & 31)] = buf[m0 * 1024 + e];
  }
  __syncthreads();

  v2f aX[4], bY[4];
#pragma unroll
  for (int kb = 0; kb < 4; ++kb) {
    const int k = 4 * kb + 2 * hi;
    bY[kb].x = Uym[row16 * 32 + k];  // B[k][n] = Uy[n][k] (diag tile)
    bY[kb].y = Uym[row16 * 32 + k + 1];
    aX[kb].x = Uxm[row16 * 32 + k];
    aX[kb].y = Uxm[row16 * 32 + k + 1];
  }

  // Stage C: contract ky (right-form, A from LDS rows, B = Uy^T tile).
  {
    const int i = wave >> 2;
    const int kt = (wave >> 1) & 1;
    const int a0 = (wave & 1) * 16;
    v8f c = {};
#pragma unroll
    for (int kb = 0; kb < 4; ++kb) {
      const int k = 4 * kb + 2 * hi;
      v2f av;
      av.x = T[i][(kt * 16 + row16) * 33 + a0 + k];
      av.y = T[i][(kt * 16 + row16) * 33 + a0 + k + 1];
      c = wmma_f32_16x16x4(av, bY[kb], c);
    }
#pragma unroll
    for (int r = 0; r < 8; ++r)
      T[i][(kt * 16 + r + 8 * hi) * 33 + a0 + row16] = c[r];
  }
  __syncthreads();

  // Stage D: contract kx (left-form, LDS row stride 33).
  {
    const int i = wave >> 2;
    const int a0 = ((wave >> 1) & 1) * 16;
    const int n0 = (wave & 1) * 16;
    v8f c = {};
#pragma unroll
    for (int kb = 0; kb < 4; ++kb) {
      const int k = 4 * kb + 2 * hi;
      v2f bv;
      bv.x = T[i][(a0 + k) * 33 + n0 + row16];
      bv.y = T[i][(a0 + k + 1) * 33 + n0 + row16];
      c = wmma_f32_16x16x4(aX[kb], bv, c);
    }
#pragma unroll
    for (int r = 0; r < 8; ++r)
      T[i][(a0 + r + 8 * hi) * 33 + n0 + row16] = c[r];
  }
  __syncthreads();

  for (int e = t; e < 2048; e += 256) {
    const int i = e >> 10;
    const int e1 = e & 1023;
    buf[m0 * 1024 + e] = T[i][(e1 >> 5) * 33 + (e1 & 31)];
  }
}

// ---------------------------------------------------------------------------
// out[b] = u @ rho[b] @ u^T, u = kron(Ux, Uy), both 32x32 block-diag(8x8 x4).
// Two passes total: fused_left (rho -> out) and fused_right (in-place).
// ---------------------------------------------------------------------------
extern "C" void kernel_launch(void* const* d_in, const int* in_sizes, int n_in,
                              void* d_out, int out_size, void* d_ws,
                              size_t ws_size, hipStream_t stream) {
  const float* rho = (const float*)d_in[0];
  const float* phi_x = (const float*)d_in[1];
  const float* phi_y = (const float*)d_in[2];
  float* out = (float*)d_out;
  float* U = (float*)d_ws;  // 8 KB: Ux then Uy
  float* Ux = U;
  float* Uy = U + 1024;

  build_unitaries<<<1, 256, 0, stream>>>(phi_x, phi_y, U);
  // 16 b x 4 jx-blk x 4 jy-blk x 16 col-chunks = 4096 blocks
  fused_left<<<4096, 256, 0, stream>>>(rho, out, Ux, Uy);
  // 16384 rows / 2 per block = 8192 blocks
  fused_right<<<8192, 256, 0, stream>>>(out, Ux, Uy);
}